// StructuralBlocDiagBoxHead_67680094651009
// MI455X (gfx1250) — compile-verified
//
#include <hip/hip_runtime.h>

// ---------------------------------------------------------------------------
// Fused block-diagonal 2-layer MLP for MI455X (gfx1250, wave32, WMMA).
//
//   out = relu( relu( x @ W1[n]^T + b1 ) @ W2[n]^T + b2 )   per block n
//
// Block-diagonality => each of the 8 column blocks is independent through
// BOTH layers, so we fuse and never write the intermediate h (134 MB) to HBM.
// HBM traffic ~285 MB -> ~12 us floor at 23.3 TB/s. Math is exact f32 via
// V_WMMA_F32_16X16X4_F32 (2048 FLOP/instr), 4096 WMMAs per wave.
// ---------------------------------------------------------------------------

typedef __attribute__((ext_vector_type(2))) float v2f;
typedef __attribute__((ext_vector_type(4))) float v4f;
typedef __attribute__((ext_vector_type(8))) float v8f;

#define NB     8
#define BIN    512
#define BOUT   512
#define DMODEL (NB * BIN)        // 4096
#define ROWS   8192
#define M_TILE 64                // rows per workgroup
#define LDS_STRIDE (BIN + 4)     // 516: row stride pad -> 4-bank skew, no conflicts
#define SMEM_BYTES (M_TILE * LDS_STRIDE * 4)   // 132096 B, 2 WGs fit per 320KB WGP

__global__ __launch_bounds__(256, 1)
void fused_blockdiag_mlp(const float* __restrict__ x,
                         const float* __restrict__ W1,
                         const float* __restrict__ b1,
                         const float* __restrict__ W2,
                         const float* __restrict__ b2,
                         float* __restrict__ out)
{
    extern __shared__ float smem[];          // [M_TILE][LDS_STRIDE], X then reused as H

    const int blk    = blockIdx.y;           // diagonal block 0..7
    const int row0   = blockIdx.x * M_TILE;  // first row of this tile
    const int tid    = threadIdx.x;
    const int wave   = tid >> 5;             // 0..7 -> 64-column group
    const int lane   = tid & 31;
    const int laneN  = lane & 15;            // N / M column within fragment
    const int laneHi = lane >> 4;            // 0/1: K-half selector (ISA A/B layout)
    const int oBase  = wave * 64;            // this wave's output-column base (0..448)

    // ---- stage X tile (64 x 512 f32) into LDS, coalesced b128 loads ----
    // x rows are read exactly once across the whole kernel -> nontemporal.
    {
        const float* xblk = x + (size_t)row0 * DMODEL + blk * BIN;
        for (int i = tid; i < M_TILE * (BIN / 4); i += 256) {
            const int r = i >> 7;            // / 128 float4 per row
            const int c = (i & 127) << 2;    // col in floats
            const v4f* src = (const v4f*)(xblk + (size_t)r * DMODEL + c);
            const v4f v = __builtin_nontemporal_load(src);
            *(v4f*)(smem + r * LDS_STRIDE + c) = v;
        }
    }
    __syncthreads();

    const v8f vzero = {0.f, 0.f, 0.f, 0.f, 0.f, 0.f, 0.f, 0.f};

    // ======================= Layer 1: h = relu(X @ W1^T + b1) ==============
    v8f acc[4][4];
    #pragma unroll
    for (int mi = 0; mi < 4; ++mi)
        #pragma unroll
        for (int ni = 0; ni < 4; ++ni) acc[mi][ni] = vzero;

    {
        const float* Wn = W1 + (size_t)blk * BOUT * BIN;  // [512][512] row-major (o, c)
        #pragma unroll 2
        for (int k = 0; k < BIN; k += 4) {
            v2f a[4], bm[4];
            #pragma unroll
            for (int mi = 0; mi < 4; ++mi) {          // A: 16x4 f32, VGPRj: K=j+2*laneHi
                const int m = mi * 16 + laneN;
                a[mi] = *(const v2f*)(smem + m * LDS_STRIDE + k + 2 * laneHi);
            }
            #pragma unroll
            for (int ni = 0; ni < 4; ++ni) {          // B: 4x16, B(k,o)=W[o][k] contig in k
                const int o = oBase + ni * 16 + laneN;
                bm[ni] = *(const v2f*)(Wn + (size_t)o * BIN + k + 2 * laneHi);
            }
            #pragma unroll
            for (int mi = 0; mi < 4; ++mi)
                #pragma unroll
                for (int ni = 0; ni < 4; ++ni)
                    acc[mi][ni] = __builtin_amdgcn_wmma_f32_16x16x4_f32(
                        false, a[mi], false, bm[ni], (short)0, acc[mi][ni],
                        false, false);
        }
    }

    __syncthreads();   // all waves done READING X -> safe to overwrite LDS with H

    // ---- epilogue 1: bias + ReLU, write H into the (reused) LDS buffer ----
    #pragma unroll
    for (int ni = 0; ni < 4; ++ni) {
        const float bv = b1[blk * BOUT + oBase + ni * 16 + laneN];
        #pragma unroll
        for (int mi = 0; mi < 4; ++mi) {
            #pragma unroll
            for (int r = 0; r < 8; ++r) {
                const int m = mi * 16 + r + 8 * laneHi;   // C/D layout: VGPRr -> M=r(+8)
                float v = acc[mi][ni][r] + bv;
                smem[m * LDS_STRIDE + oBase + ni * 16 + laneN] = v > 0.f ? v : 0.f;
            }
        }
    }
    __syncthreads();   // H fully visible to all waves

    // ======================= Layer 2: out = relu(H @ W2^T + b2) ============
    #pragma unroll
    for (int mi = 0; mi < 4; ++mi)
        #pragma unroll
        for (int ni = 0; ni < 4; ++ni) acc[mi][ni] = vzero;

    {
        const float* Wn = W2 + (size_t)blk * BOUT * BIN;
        #pragma unroll 2
        for (int k = 0; k < BIN; k += 4) {
            v2f a[4], bm[4];
            #pragma unroll
            for (int mi = 0; mi < 4; ++mi) {
                const int m = mi * 16 + laneN;
                a[mi] = *(const v2f*)(smem + m * LDS_STRIDE + k + 2 * laneHi);
            }
            #pragma unroll
            for (int ni = 0; ni < 4; ++ni) {
                const int o = oBase + ni * 16 + laneN;
                bm[ni] = *(const v2f*)(Wn + (size_t)o * BIN + k + 2 * laneHi);
            }
            #pragma unroll
            for (int mi = 0; mi < 4; ++mi)
                #pragma unroll
                for (int ni = 0; ni < 4; ++ni)
                    acc[mi][ni] = __builtin_amdgcn_wmma_f32_16x16x4_f32(
                        false, a[mi], false, bm[ni], (short)0, acc[mi][ni],
                        false, false);
        }
    }

    // ---- epilogue 2: bias + ReLU, nontemporal store to global out ----
    // out is write-once streaming data: NT keeps it from displacing the
    // weight working set (16.8 MB, fully L2-resident) in the 192 MB L2.
    #pragma unroll
    for (int ni = 0; ni < 4; ++ni) {
        const int col = oBase + ni * 16 + laneN;
        const float bv = b2[blk * BOUT + col];
        #pragma unroll
        for (int mi = 0; mi < 4; ++mi) {
            #pragma unroll
            for (int r = 0; r < 8; ++r) {
                const int m = mi * 16 + r + 8 * laneHi;
                float v = acc[mi][ni][r] + bv;
                __builtin_nontemporal_store(v > 0.f ? v : 0.f,
                    out + (size_t)(row0 + m) * DMODEL + blk * BIN + col);
            }
        }
    }
}

extern "C" void kernel_launch(void* const* d_in, const int* in_sizes, int n_in,
                              void* d_out, int out_size, void* d_ws, size_t ws_size,
                              hipStream_t stream) {
    (void)in_sizes; (void)n_in; (void)out_size; (void)d_ws; (void)ws_size;
    const float* x  = (const float*)d_in[0];
    const float* W1 = (const float*)d_in[1];
    const float* b1 = (const float*)d_in[2];
    const float* W2 = (const float*)d_in[3];
    const float* b2 = (const float*)d_in[4];
    float* out = (float*)d_out;

    // 129 KB dynamic LDS (> default 64 KB cap) -> raise the attribute once.
    static bool attr_set = false;
    if (!attr_set) {
        (void)hipFuncSetAttribute((const void*)fused_blockdiag_mlp,
                                  hipFuncAttributeMaxDynamicSharedMemorySize,
                                  SMEM_BYTES);
        attr_set = true;
    }

    dim3 grid(ROWS / M_TILE, NB);   // 128 x 8 workgroups
    dim3 block(256);                // 8 waves
    fused_blockdiag_mlp<<<grid, block, SMEM_BYTES, stream>>>(x, W1, b1, W2, b2, out);
}